// HierarchicalCellEncoder_27711128994331
// MI455X (gfx1250) — compile-verified
//
#include <hip/hip_runtime.h>
#include <hip/hip_bf16.h>

typedef __attribute__((ext_vector_type(2))) float v2f;
typedef __attribute__((ext_vector_type(8))) float v8f;

#define DHID 256  // all GEMMs have 256 output columns; K is 256 per input panel

// ---------------------------------------------------------------------------
// Gather + masked mean: out[c, :] = sum_k mask[c,k]*src[idx[c,k], :] / max(cnt,1)
// One block of 256 threads per output row; thread t owns column t.
// ---------------------------------------------------------------------------
__global__ __launch_bounds__(256)
void masked_mean_kernel(const float* __restrict__ src,   // [*, 256]
                        const int*   __restrict__ idx,   // [C, K]
                        const float* __restrict__ mask,  // [C, K]
                        float*       __restrict__ out,   // [C, 256]
                        int C, int K)
{
    const int c = blockIdx.x;
    if (c >= C) return;
    const int t = threadIdx.x;  // 0..255 -> column
    float s = 0.0f, cnt = 0.0f;
    for (int k = 0; k < K; ++k) {
        const float m = mask[(size_t)c * K + k];   // block-uniform
        const int   i = idx [(size_t)c * K + k];   // block-uniform
        cnt += m;
        if (m != 0.0f)
            s += src[(size_t)i * DHID + t] * m;    // coalesced 1KB row gather
    }
    const float r = (cnt > 0.0f) ? s / fmaxf(cnt, 1.0f) : 0.0f;
    out[(size_t)c * DHID + t] = r;
}

// ---------------------------------------------------------------------------
// Repack W [Kdim,256] row-major into WMMA B-fragment order so that each
// lane's (k,k+1) value pair is a contiguous b64 and consecutive lanes are
// contiguous:
//   Wpk[ kstep*1024 + (nt*32 + lane)*2 + {0,1} ]
//     = W[ kstep*4 + (lane>>4)*2 + {0,1} ][ nt*16 + (lane&15) ]
// One-time cost (<= 512KB per matrix), amortized over all GEMM blocks.
// ---------------------------------------------------------------------------
__global__ __launch_bounds__(256)
void repack_w_kernel(const float* __restrict__ W, float* __restrict__ Wpk, int Kdim)
{
    const int tid   = (int)blockIdx.x * 256 + (int)threadIdx.x;
    const int total = (Kdim / 4) * 16 * 32;        // one thread per (kstep, nt, lane)
    if (tid >= total) return;
    const int lane = tid & 31;
    const int nt   = (tid >> 5) & 15;
    const int ks   = tid >> 9;
    const int col  = nt * 16 + (lane & 15);
    const int kr   = ks * 4 + (lane >> 4) * 2;
    Wpk[(size_t)tid * 2 + 0] = W[(size_t)kr * DHID + col];
    Wpk[(size_t)tid * 2 + 1] = W[(size_t)(kr + 1) * DHID + col];
}

// ---------------------------------------------------------------------------
// Y[M,256] = act( [Xa | Xb] @ W + bias ),  W given pre-packed (see above),
// covering Kdim = 256 (Xb==nullptr) or 512 (Xb feeds packed ksteps [64,128)).
// One wave computes a 16-row x 256-col strip with v_wmma_f32_16x16x4_f32
// (exact fp32). Block = 256 threads = 8 waves = 128 rows.
//
// f32 16x16x4 operand layouts (ISA 7.12.2):
//   A (16x4): lanes 0-15: M=lane, v0=K0, v1=K1 ; lanes 16-31: M=lane-16, v0=K2, v1=K3
//   B (4x16): lanes 0-15: N=lane, v0=K0, v1=K1 ; lanes 16-31: N=lane-16, v0=K2, v1=K3
//   C/D:      VGPR j: lanes 0-15 -> row j, lanes 16-31 -> row j+8; col = lane&15
// ---------------------------------------------------------------------------
template <bool RELU>
__global__ __launch_bounds__(256)
void gemm_wmma_f32(const float* __restrict__ Xa,   // [M,256]
                   const float* __restrict__ Xb,   // [M,256] or nullptr
                   const float* __restrict__ Wpk,  // packed [Kdim/4][16][32][2]
                   const float* __restrict__ bias, // [256]
                   float*       __restrict__ Y,    // [M,256]
                   int M)
{
    const int wave = threadIdx.x >> 5;
    const int lane = threadIdx.x & 31;
    const int m0   = ((int)blockIdx.x * 8 + wave) * 16;
    if (m0 >= M) return;                 // wave-uniform: EXEC stays all-ones

    const int rsel = lane & 15;          // row-in-tile (A) / col-in-tile (B/D)
    const int hi   = lane >> 4;          // 0 or 1
    const int koff = hi * 2;             // K sub-offset for this lane half

    v8f acc[16];
    const v8f zero = {};
#pragma unroll
    for (int nt = 0; nt < 16; ++nt) acc[nt] = zero;

    const float* Xs[2] = { Xa, Xb };
    for (int pass = 0; pass < 2; ++pass) {
        const float* Xp = Xs[pass];
        if (!Xp) continue;
        const float* ar = Xp + (size_t)(m0 + rsel) * DHID;     // this lane's A row

        for (int k0 = 0; k0 < DHID; k0 += 4) {
            const v2f afrag = *(const v2f*)(ar + k0 + koff);   // b64, 8B-aligned
            const float* bb = Wpk + (size_t)(pass * 64 + (k0 >> 2)) * 1024
                                  + (size_t)lane * 2;
#pragma unroll
            for (int nt = 0; nt < 16; ++nt) {
                const v2f bfrag = *(const v2f*)(bb + nt * 64); // coalesced b64
                acc[nt] = __builtin_amdgcn_wmma_f32_16x16x4_f32(
                    /*neg_a=*/false, afrag, /*neg_b=*/false, bfrag,
                    /*c_mod=*/(short)0, acc[nt],
                    /*reuse_a=*/false, /*reuse_b=*/false);
            }
        }
    }

    // Epilogue: bias (+ ReLU), store per documented C/D layout.
#pragma unroll
    for (int nt = 0; nt < 16; ++nt) {
        const int   col = nt * 16 + rsel;
        const float bv  = bias[col];
#pragma unroll
        for (int j = 0; j < 8; ++j) {
            float v = acc[nt][j] + bv;
            if (RELU) v = fmaxf(v, 0.0f);
            Y[(size_t)(m0 + j + hi * 8) * DHID + col] = v;
        }
    }
}

// ---------------------------------------------------------------------------
// Host-side orchestration.
// ---------------------------------------------------------------------------
extern "C" void kernel_launch(void* const* d_in, const int* in_sizes, int n_in,
                              void* d_out, int out_size, void* d_ws, size_t ws_size,
                              hipStream_t stream)
{
    const float* X    = (const float*)d_in[0];
    const int*   d1ci = (const int*)  d_in[1];
    const int*   d1bi = (const int*)  d_in[2];
    const int*   d2ci = (const int*)  d_in[3];
    const int*   d2bi = (const int*)  d_in[4];
    const float* d1cm = (const float*)d_in[5];
    const float* d1bm = (const float*)d_in[6];
    const float* d2cm = (const float*)d_in[7];
    const float* d2bm = (const float*)d_in[8];
    const float* W0   = (const float*)d_in[9];
    const float* b0   = (const float*)d_in[10];
    const float* W1a  = (const float*)d_in[11];
    const float* b1a  = (const float*)d_in[12];
    const float* W1b  = (const float*)d_in[13];
    const float* b1b  = (const float*)d_in[14];
    const float* W2a  = (const float*)d_in[15];
    const float* b2a  = (const float*)d_in[16];
    const float* W2b  = (const float*)d_in[17];
    const float* b2b  = (const float*)d_in[18];
    const float* Wout = (const float*)d_in[19];
    const float* bout = (const float*)d_in[20];

    const int D  = DHID;
    const int K1 = 16, K2 = 32, B2 = 8;
    const int N  = in_sizes[0] / D;    // 100000
    const int C1 = in_sizes[1] / K1;   // 50000
    const int C2 = in_sizes[3] / K2;   // 10000

    // Scratch layout (floats):
    //   P0: N*256     hidden0; later reused as d1_direct + d1_bnd, then h1b
    //   P1: C1*256    mean1; later h1a
    //   P2: 3*C2*256  mean2 / d2_direct / d2_bnd, reused for h2a / h2b
    //   PW: packed weights (W0, Wout, W1a, W1b, W2a, W2b)
    float* P0 = (float*)d_ws;
    float* P1 = P0 + (size_t)N  * D;
    float* P2 = P1 + (size_t)C1 * D;
    float* PW = P2 + (size_t)3 * C2 * D;

    float* w0pk   = PW;                       // 256*256
    float* woutpk = w0pk   + (size_t)D * D;   // 256*256
    float* w1apk  = woutpk + (size_t)D * D;   // 512*256
    float* w1bpk  = w1apk  + (size_t)2 * D * D;
    float* w2apk  = w1bpk  + (size_t)D * D;   // 512*256
    float* w2bpk  = w2apk  + (size_t)2 * D * D;

    float* d1_direct = P0;
    float* d1_bnd    = P0 + (size_t)C1 * D;
    float* h1b       = P0;
    float* mean1     = P1;
    float* h1a       = P1;
    float* mean2     = P2;
    float* d2_direct = P2 + (size_t)C2 * D;
    float* d2_bnd    = P2 + (size_t)2 * C2 * D;
    float* h2a       = P2;
    float* h2b       = P2 + (size_t)C2 * D;

    float* emb0 = (float*)d_out;
    float* emb1 = emb0 + (size_t)N  * D;
    float* emb2 = emb1 + (size_t)C1 * D;

    const dim3 blk(256);
    auto gg = [](int M) { return dim3((unsigned)((M + 127) / 128)); };
    auto rg = [](int Kdim) { return dim3((unsigned)((Kdim * 128 + 255) / 256)); };

    // One-time weight repack into WMMA B-fragment order (tiny; L2-resident after).
    repack_w_kernel<<<rg(256), blk, 0, stream>>>(W0,   w0pk,   256);
    repack_w_kernel<<<rg(256), blk, 0, stream>>>(Wout, woutpk, 256);
    repack_w_kernel<<<rg(512), blk, 0, stream>>>(W1a,  w1apk,  512);
    repack_w_kernel<<<rg(256), blk, 0, stream>>>(W1b,  w1bpk,  256);
    repack_w_kernel<<<rg(512), blk, 0, stream>>>(W2a,  w2apk,  512);
    repack_w_kernel<<<rg(256), blk, 0, stream>>>(W2b,  w2bpk,  256);

    // dim 0: emb0 = proj(lin0(X))
    gemm_wmma_f32<false><<<gg(N), blk, 0, stream>>>(X, nullptr, w0pk, b0, P0 /*hidden0*/, N);
    gemm_wmma_f32<false><<<gg(N), blk, 0, stream>>>(P0, nullptr, woutpk, bout, emb0, N);

    // dim 1
    masked_mean_kernel<<<dim3((unsigned)C1), blk, 0, stream>>>(X, d1ci, d1cm, mean1, C1, K1);
    gemm_wmma_f32<false><<<gg(C1), blk, 0, stream>>>(mean1, nullptr, w0pk, b0, d1_direct, C1);
    masked_mean_kernel<<<dim3((unsigned)C1), blk, 0, stream>>>(emb0, d1bi, d1bm, d1_bnd, C1, K1);
    gemm_wmma_f32<true ><<<gg(C1), blk, 0, stream>>>(d1_direct, d1_bnd, w1apk, b1a, h1a, C1);
    gemm_wmma_f32<false><<<gg(C1), blk, 0, stream>>>(h1a, nullptr, w1bpk, b1b, h1b, C1);
    gemm_wmma_f32<false><<<gg(C1), blk, 0, stream>>>(h1b, nullptr, woutpk, bout, emb1, C1);

    // dim 2
    masked_mean_kernel<<<dim3((unsigned)C2), blk, 0, stream>>>(X, d2ci, d2cm, mean2, C2, K2);
    gemm_wmma_f32<false><<<gg(C2), blk, 0, stream>>>(mean2, nullptr, w0pk, b0, d2_direct, C2);
    masked_mean_kernel<<<dim3((unsigned)C2), blk, 0, stream>>>(emb1, d2bi, d2bm, d2_bnd, C2, B2);
    gemm_wmma_f32<true ><<<gg(C2), blk, 0, stream>>>(d2_direct, d2_bnd, w2apk, b2a, h2a, C2);
    gemm_wmma_f32<false><<<gg(C2), blk, 0, stream>>>(h2a, nullptr, w2bpk, b2b, h2b, C2);
    gemm_wmma_f32<false><<<gg(C2), blk, 0, stream>>>(h2b, nullptr, woutpk, bout, emb2, C2);
}